// MashSSM_62234076119901
// MI455X (gfx1250) — compile-verified
//
#include <hip/hip_runtime.h>
#include <hip/hip_bf16.h>

// ---------------------------------------------------------------------------
// MashSSM forward, collapsed form (mathematically exact):
//   softmax over a single context token == 1  =>  lat = v (broadcast over L)
//   => entire Mamba trunk / point-embed / time-embed are dead code,
//   => fuse+out projection is rank-1 in the sequence dimension.
// Surviving compute: 4 GEMMs with M=16 (one WMMA tile), done in full f32 with
// v_wmma_f32_16x16x4_f32 so weights stream straight from HBM (no downcast
// pass): total traffic ~14 MB, exact f32 math matching the reference.
// ---------------------------------------------------------------------------

typedef __attribute__((ext_vector_type(2))) float v2f;
typedef __attribute__((ext_vector_type(8))) float v8f;

#define DMODEL 768
#define BATCH  16
#define NPTS   400
#define PDIM_  22
#define LSEQ   402   // te + cond + 400 points

// ------------------------- cond = cat_emb[idx] * (1-p) ---------------------
__global__ void cond_gather_kernel(const float* __restrict__ cat_emb,
                                   const int* __restrict__ cond,
                                   const int* __restrict__ dropp,
                                   float* __restrict__ out) {
  int b = blockIdx.x;
  float scale = 1.0f - (float)dropp[0];
  int idx = cond[b];
  for (int d = threadIdx.x; d < DMODEL; d += blockDim.x)
    out[b * DMODEL + d] = cat_emb[idx * DMODEL + d] * scale;
}

// ------------------------- layernorm rows (16 x 768) -----------------------
__global__ void layernorm16_kernel(const float* __restrict__ X,
                                   const float* __restrict__ w,
                                   const float* __restrict__ b,
                                   float* __restrict__ out) {
  __shared__ float red[256];
  int row = blockIdx.x, tid = threadIdx.x;
  const float* x = X + row * DMODEL;
  float s = 0.f;
  for (int d = tid; d < DMODEL; d += 256) s += x[d];
  red[tid] = s; __syncthreads();
  for (int o = 128; o > 0; o >>= 1) { if (tid < o) red[tid] += red[tid + o]; __syncthreads(); }
  float mean = red[0] * (1.0f / DMODEL);
  __syncthreads();
  float v = 0.f;
  for (int d = tid; d < DMODEL; d += 256) { float t = x[d] - mean; v += t * t; }
  red[tid] = v; __syncthreads();
  for (int o = 128; o > 0; o >>= 1) { if (tid < o) red[tid] += red[tid + o]; __syncthreads(); }
  float inv = rsqrtf(red[0] * (1.0f / DMODEL) + 1e-5f);
  for (int d = tid; d < DMODEL; d += 256)
    out[row * DMODEL + d] = (x[d] - mean) * inv * w[d] + b[d];
}

// ---------------- WMMA f32 GEMM: C[16,N] = A[16,K] @ W[N,K]^T --------------
// One wave per 16x16 output tile; K consumed 4/step with
// v_wmma_f32_16x16x4_f32 (full f32).  K % 4 == 0, N % 16 == 0.
// 32-bit A 16x4 layout: lanes 0-15 M=lane {K=0,1}; lanes 16-31 {K=2,3}.
__global__ __launch_bounds__(32)
void wmma_gemv16_f32_kernel(const float* __restrict__ A,      // [16,K]
                            const float* __restrict__ W,      // [N,K] (row = output col)
                            const float* __restrict__ bias,   // [N] or null
                            const float* __restrict__ addsrc, // [16,N] residual or null
                            float* __restrict__ Cf,           // [16,N]
                            int N, int K) {
  const int lane = threadIdx.x;            // 0..31 (wave32)
  const int m    = lane & 15;              // A row this lane carries
  const int kb   = (lane >> 4) << 1;       // 0 or 2 (K sub-block)
  const int n    = (blockIdx.x << 4) + (lane & 15);

  const float* arow = A + (size_t)m * K;
  const float* wrow = W + (size_t)n * K;

  v8f acc = {};
#pragma unroll 16
  for (int k0 = 0; k0 < K; k0 += 4) {
    v2f a = *(const v2f*)(arow + k0 + kb);
    v2f b = *(const v2f*)(wrow + k0 + kb);
    if ((k0 & 63) == 0)                               // one prefetch / 256B
      __builtin_prefetch(wrow + k0 + 256, 0, 1);      // global_prefetch_b8
    acc = __builtin_amdgcn_wmma_f32_16x16x4_f32(
        false, a, false, b, (short)0, acc, false, false);
  }

  const int mbase = (lane >> 4) << 3;   // VGPR r holds M = r (+8 for hi lanes)
#pragma unroll
  for (int r = 0; r < 8; ++r) {
    int mo = mbase + r;
    float v = acc[r];
    if (bias)   v += bias[n];
    if (addsrc) v += addsrc[mo * N + n];
    Cf[mo * N + n] = v;
  }
}

// ------------------------- GEGLU: a * gelu_tanh(gate) ----------------------
__global__ void geglu_kernel(const float* __restrict__ ff1out,  // [16, 1536]
                             float* __restrict__ g_out) {       // [16, 768]
  int i = blockIdx.x * blockDim.x + threadIdx.x;
  if (i >= BATCH * DMODEL) return;
  int m = i / DMODEL, d = i % DMODEL;
  float a = ff1out[m * 2 * DMODEL + d];
  float g = ff1out[m * 2 * DMODEL + DMODEL + d];
  float t = tanhf(0.7978845608028654f * (g + 0.044715f * g * g * g));
  g_out[i] = a * 0.5f * g * (1.0f + t);
}

// --------- S[n] = sum_m fuse_w[n,m];  rowsum[p] = sum_d out_w[p,d] ---------
__global__ void sums_kernel(const float* __restrict__ fuse_w,
                            const float* __restrict__ out_w,
                            float* __restrict__ S, float* __restrict__ rowsum) {
  int tid = threadIdx.x;
  if (tid < NPTS) {
    float s = 0.f;
    for (int m = 0; m < LSEQ; ++m) s += fuse_w[tid * LSEQ + m];
    S[tid] = s;
  } else if (tid < NPTS + PDIM_) {
    int p = tid - NPTS;
    float s = 0.f;
    for (int d = 0; d < DMODEL; ++d) s += out_w[p * DMODEL + d];
    rowsum[p] = s;
  }
}

// ------------------------- t1[b,p] = latv2[b] . out_w[p] -------------------
__global__ void dot22_kernel(const float* __restrict__ latv2,
                             const float* __restrict__ out_w,
                             float* __restrict__ t1) {
  __shared__ float red[256];
  int b = blockIdx.x / PDIM_, p = blockIdx.x % PDIM_;
  int tid = threadIdx.x;
  float s = 0.f;
  for (int d = tid; d < DMODEL; d += 256)
    s += latv2[b * DMODEL + d] * out_w[p * DMODEL + d];
  red[tid] = s; __syncthreads();
  for (int o = 128; o > 0; o >>= 1) { if (tid < o) red[tid] += red[tid + o]; __syncthreads(); }
  if (tid == 0) t1[b * PDIM_ + p] = red[0];
}

// ---- out[b,n,p] = S[n]*t1[b,p] + fuse_b[n]*rowsum[p] + out_b[p] -----------
__global__ void final_out_kernel(const float* __restrict__ S,
                                 const float* __restrict__ rowsum,
                                 const float* __restrict__ t1,
                                 const float* __restrict__ fuse_b,
                                 const float* __restrict__ out_b,
                                 float* __restrict__ out) {
  int i = blockIdx.x * blockDim.x + threadIdx.x;
  if (i >= BATCH * NPTS * PDIM_) return;
  int p = i % PDIM_;
  int n = (i / PDIM_) % NPTS;
  int b = i / (PDIM_ * NPTS);
  out[i] = S[n] * t1[b * PDIM_ + p] + fuse_b[n] * rowsum[p] + out_b[p];
}

// ---------------------------------------------------------------------------
extern "C" void kernel_launch(void* const* d_in, const int* in_sizes, int n_in,
                              void* d_out, int out_size, void* d_ws, size_t ws_size,
                              hipStream_t stream) {
  (void)in_sizes; (void)n_in; (void)out_size; (void)ws_size;

  // setup_inputs() flattening order:
  // 0 mash_params, 1 condition, 2 t, 3 condition_drop_prob,
  // 4..15 point-embed params, 16..19 time MLP, 20 cat_emb,
  // 21..140 layers (12 x 10), 141 norm_f_w, 142 ca_norm_w, 143 ca_norm_b,
  // 144 ca_ctx_norm_w, 145 ca_ctx_norm_b, 146 q_w, 147 kv_w, 148 ao_w,
  // 149 ao_b, 150 ff_norm_w, 151 ff_norm_b, 152 ff1_w, 153 ff1_b,
  // 154 ff2_w, 155 ff2_b, 156 fuse_w, 157 fuse_b, 158 out_w, 159 out_b
  const int*   condition = (const int*)  d_in[1];
  const int*   dropp     = (const int*)  d_in[3];
  const float* cat_emb   = (const float*)d_in[20];
  const float* ctxn_w    = (const float*)d_in[144];
  const float* ctxn_b    = (const float*)d_in[145];
  const float* kv_w      = (const float*)d_in[147];
  const float* ao_w      = (const float*)d_in[148];
  const float* ao_b      = (const float*)d_in[149];
  const float* ffn_w     = (const float*)d_in[150];
  const float* ffn_b     = (const float*)d_in[151];
  const float* ff1_w     = (const float*)d_in[152];
  const float* ff1_b     = (const float*)d_in[153];
  const float* ff2_w     = (const float*)d_in[154];
  const float* ff2_b     = (const float*)d_in[155];
  const float* fuse_w    = (const float*)d_in[156];
  const float* fuse_b    = (const float*)d_in[157];
  const float* out_w     = (const float*)d_in[158];
  const float* out_b     = (const float*)d_in[159];
  float* out = (float*)d_out;

  // ---- workspace carve-up (256B aligned slabs, ~436 KB total) ----
  char* ws = (char*)d_ws;
  size_t off = 0;
  auto alloc = [&](size_t bytes) {
    void* p = (void*)(ws + off);
    off += (bytes + 255) & ~(size_t)255;
    return p;
  };
  float* condb  = (float*)alloc((size_t)BATCH * DMODEL * 4);
  float* ctx    = (float*)alloc((size_t)BATCH * DMODEL * 4);
  float* vbuf   = (float*)alloc((size_t)BATCH * DMODEL * 4);
  float* latv   = (float*)alloc((size_t)BATCH * DMODEL * 4);
  float* xf     = (float*)alloc((size_t)BATCH * DMODEL * 4);
  float* ff1o   = (float*)alloc((size_t)BATCH * 2 * DMODEL * 4);
  float* gbuf   = (float*)alloc((size_t)BATCH * DMODEL * 4);
  float* latv2  = (float*)alloc((size_t)BATCH * DMODEL * 4);
  float* Ssum   = (float*)alloc((size_t)NPTS * 4);
  float* rowsum = (float*)alloc((size_t)PDIM_ * 4);
  float* t1     = (float*)alloc((size_t)BATCH * PDIM_ * 4);

  // ---- cond -> layernorm -> ctx ----
  cond_gather_kernel<<<BATCH, 256, 0, stream>>>(cat_emb, condition, dropp, condb);
  layernorm16_kernel<<<BATCH, 256, 0, stream>>>(condb, ctxn_w, ctxn_b, ctx);

  // ---- v = ctx @ kv_w[768:].T            (att == 1 exactly; q/k are dead) --
  wmma_gemv16_f32_kernel<<<DMODEL / 16, 32, 0, stream>>>(
      ctx, kv_w + (size_t)DMODEL * DMODEL, nullptr, nullptr, vbuf, DMODEL, DMODEL);

  // ---- lat = v @ ao_w.T + ao_b  (constant over sequence dim) ----
  wmma_gemv16_f32_kernel<<<DMODEL / 16, 32, 0, stream>>>(
      vbuf, ao_w, ao_b, nullptr, latv, DMODEL, DMODEL);

  // ---- xf = layernorm(lat) ----
  layernorm16_kernel<<<BATCH, 256, 0, stream>>>(latv, ffn_w, ffn_b, xf);

  // ---- ff1 ----
  wmma_gemv16_f32_kernel<<<2 * DMODEL / 16, 32, 0, stream>>>(
      xf, ff1_w, ff1_b, nullptr, ff1o, 2 * DMODEL, DMODEL);

  // ---- a * gelu(gate) ----
  geglu_kernel<<<(BATCH * DMODEL + 255) / 256, 256, 0, stream>>>(ff1o, gbuf);

  // ---- lat2 = lat + geglu @ ff2_w.T + ff2_b ----
  wmma_gemv16_f32_kernel<<<DMODEL / 16, 32, 0, stream>>>(
      gbuf, ff2_w, ff2_b, latv, latv2, DMODEL, DMODEL);

  // ---- rank-1 fuse + output projection ----
  sums_kernel<<<1, 448, 0, stream>>>(fuse_w, out_w, Ssum, rowsum);
  dot22_kernel<<<BATCH * PDIM_, 256, 0, stream>>>(latv2, out_w, t1);
  final_out_kernel<<<(BATCH * NPTS * PDIM_ + 255) / 256, 256, 0, stream>>>(
      Ssum, rowsum, t1, fuse_b, out_b, out);
}